// Pooler_19816979103972
// MI455X (gfx1250) — compile-verified
//
#include <hip/hip_runtime.h>
#include <stdint.h>

// FPN multi-scale ROIAlign for MI455X (gfx1250).
// Features (183 MB) are L2-resident; kernel is gather-bound. One workgroup =
// (ROI, 8-channel block). Wave 0 stages the clamped ROI patch (a 3D tile
// wp x hp x 8ch of the NCHW tensor) into LDS with a SINGLE Tensor Data Mover
// op (TENSOR_LOAD_TO_LDS, TENSORcnt), then all 8 waves compute 8x49 pooled
// outputs from LDS via ds_load. Oversized patches fall back to direct global
// gathers with identical math.

#define OUTD 7
#define SRD  2

constexpr int BN  = 2;     // batch
constexpr int NN  = 256;   // rois per image
constexpr int CC  = 256;   // channels
constexpr int CPB = 8;     // channels per block
constexpr int PATCH_MAX = 1536;   // floats per channel staged in LDS
constexpr int TPB = 256;          // 8 waves (wave32)

typedef uint32_t u32x4 __attribute__((ext_vector_type(4)));
typedef uint32_t u32x8 __attribute__((ext_vector_type(8)));

__device__ __forceinline__ void wait_tensorcnt0() {
#if defined(__has_builtin)
#if __has_builtin(__builtin_amdgcn_s_wait_tensorcnt)
    __builtin_amdgcn_s_wait_tensorcnt(0);
    return;
#endif
#endif
    asm volatile("s_wait_tensorcnt 0x0" ::: "memory");
}

__global__ __launch_bounds__(TPB)
void roi_align_fpn_kernel(const float* __restrict__ x0,
                          const float* __restrict__ x1,
                          const float* __restrict__ x2,
                          const float* __restrict__ x3,
                          const float* __restrict__ boxes,
                          float* __restrict__ out)
{
    __shared__ float s_patch[CPB * PATCH_MAX];   // 48 KB of the 320 KB/WGP LDS

    const int r  = blockIdx.x;          // ROI index 0..511
    const int c0 = blockIdx.y * CPB;    // first channel of this block
    const int b  = r / NN;              // batch index

    // ---- per-ROI uniform setup (scalarized by the compiler) ----
    const float bx1 = boxes[r * 4 + 0];
    const float by1 = boxes[r * 4 + 1];
    const float bx2 = boxes[r * 4 + 2];
    const float by2 = boxes[r * 4 + 3];

    // FPN level: floor(4 + log2(sqrt(area)/224 + 1e-6)) clipped to [2,5], -2
    const float area = (bx2 - bx1) * (by2 - by1);
    const float sca  = sqrtf(fmaxf(area, 0.0f));
    float lvlf = floorf(4.0f + log2f(sca * (1.0f / 224.0f) + 1e-6f));
    lvlf = fminf(fmaxf(lvlf, 2.0f), 5.0f);
    const int lvl = (int)lvlf - 2;

    int H, W;
    float sc;
    const float* feat;
    if (lvl == 0)      { feat = x0; H = 200; W = 336; sc = 0.25f;    }
    else if (lvl == 1) { feat = x1; H = 100; W = 168; sc = 0.125f;   }
    else if (lvl == 2) { feat = x2; H = 50;  W = 84;  sc = 0.0625f;  }
    else               { feat = x3; H = 25;  W = 42;  sc = 0.03125f; }

    const float x1s = bx1 * sc, y1s = by1 * sc;
    const float x2s = bx2 * sc, y2s = by2 * sc;
    const float roi_w = fmaxf(x2s - x1s, 1.0f);
    const float roi_h = fmaxf(y2s - y1s, 1.0f);
    const float bin_w = roi_w * (1.0f / OUTD);
    const float bin_h = roi_h * (1.0f / OUTD);

    // Sample grid g = ph + (s+0.5)/2 is monotone in [0.25, 6.75]; all clipped
    // taps fall in [floor(clip(min)), min(floor(clip(max))+1, dim-1)].
    const float fW1 = (float)(W - 1), fH1 = (float)(H - 1);
    const float xs_min = x1s + 0.25f * bin_w, xs_max = x1s + 6.75f * bin_w;
    const float ys_min = y1s + 0.25f * bin_h, ys_max = y1s + 6.75f * bin_h;
    const int px0 = (int)floorf(fminf(fmaxf(xs_min, 0.0f), fW1));
    const int px1 = min((int)floorf(fminf(fmaxf(xs_max, 0.0f), fW1)) + 1, W - 1);
    const int py0 = (int)floorf(fminf(fmaxf(ys_min, 0.0f), fH1));
    const int py1 = min((int)floorf(fminf(fmaxf(ys_max, 0.0f), fH1)) + 1, H - 1);
    const int wp = px1 - px0 + 1;
    const int hp = py1 - py0 + 1;
    const int patch = wp * hp;
    const bool use_lds = (wp > 0) && (hp > 0) && (patch > 0) && (patch <= PATCH_MAX);

    const size_t chan_base = (size_t)(b * CC + c0) * (size_t)H * (size_t)W;

    // ---- phase 1: ONE Tensor Data Mover op stages the wp x hp x CPB tile ----
    if (use_lds && threadIdx.x < 32) {   // wave 0 only issues the TDM op
        const uint64_t gaddr =
            (uint64_t)(uintptr_t)(feat + chan_base + (size_t)py0 * (size_t)W + (size_t)px0);
        // Low 32 bits of the generic shared pointer == LDS byte offset.
        const uint32_t lds_byte = (uint32_t)(uintptr_t)(&s_patch[0]);
        const uint64_t strideC  = (uint64_t)H * (uint64_t)W;   // dim1 stride (elements)

        // D# group 0: count=1 | lds_addr | global_addr[56:0] | type=2 (image)
        u32x4 g0;
        g0.x = 1u;
        g0.y = lds_byte;
        g0.z = (uint32_t)(gaddr & 0xFFFFFFFFu);
        g0.w = (uint32_t)((gaddr >> 32) & 0x01FFFFFFu) | (2u << 30);

        // D# group 1: data_size=4B; tensor dims huge (tile pre-clamped, no OOB);
        // tile = (wp, hp, CPB); dim0_stride = W; dim1_stride = H*W.
        u32x8 g1;
        g1.s0 = (2u << 16);                                   // data_size=2 (4B)
        g1.s1 = 0xFFFF0000u;                                  // tensor_dim0[15:0]=0xFFFF
        g1.s2 = 0x7FFFu | 0xFFFF0000u;                        // dim0 hi | dim1 lo
        g1.s3 = 0x7FFFu | ((uint32_t)wp << 16);               // dim1 hi | tile_dim0
        g1.s4 = (uint32_t)hp | ((uint32_t)CPB << 16);         // tile_dim1 | tile_dim2
        g1.s5 = (uint32_t)W;                                  // dim0_stride[31:0]
        g1.s6 = ((uint32_t)(strideC & 0xFFFFu) << 16);        // dim0_str[47:32]=0 | dim1_str[15:0]
        g1.s7 = (uint32_t)(strideC >> 16);                    // dim1_stride[47:16]

        // D# group 2: tensor_dim2 huge; tile_dim3=0 (3D tile only)
        u32x4 g2;
        g2.x = 0x7FFFFFFFu;
        g2.y = 0u;
        g2.z = 0u;
        g2.w = 0u;

        u32x4 g3;
        g3.x = 0u; g3.y = 0u; g3.z = 0u; g3.w = 0u;

        asm volatile("tensor_load_to_lds %0, %1, %2, %3"
                     :
                     : "s"(g0), "s"(g1), "s"(g2), "s"(g3)
                     : "memory");
        wait_tensorcnt0();
    }
    __syncthreads();

    // ---- phase 2: bilinear pooling, CPB*49 outputs per block ----
    for (int item = threadIdx.x; item < CPB * (OUTD * OUTD); item += TPB) {
        const int c   = item / (OUTD * OUTD);
        const int bin = item - c * (OUTD * OUTD);
        const int ph  = bin / OUTD;
        const int pw  = bin - ph * OUTD;

        float acc = 0.0f;
        if (use_lds) {
            const int cbase = c * patch;
#pragma unroll
            for (int sy = 0; sy < SRD; ++sy) {
                const float gy = (float)ph + ((float)sy + 0.5f) * 0.5f;
                const float yv = y1s + gy * bin_h;
                const bool  vy = (yv >= -1.0f) && (yv <= (float)H);
                const float yc = fminf(fmaxf(yv, 0.0f), fH1);
                const int   yl = (int)floorf(yc);
                const int   yh = min(yl + 1, H - 1);
                const float fy = yc - (float)yl;
                const int   r0 = cbase + (yl - py0) * wp - px0;
                const int   r1 = cbase + (yh - py0) * wp - px0;
#pragma unroll
                for (int sx = 0; sx < SRD; ++sx) {
                    const float gx = (float)pw + ((float)sx + 0.5f) * 0.5f;
                    const float xv = x1s + gx * bin_w;
                    const bool  vx = (xv >= -1.0f) && (xv <= (float)W);
                    const float xc = fminf(fmaxf(xv, 0.0f), fW1);
                    const int   xl = (int)floorf(xc);
                    const int   xh = min(xl + 1, W - 1);
                    const float fx = xc - (float)xl;
                    // direct s_patch[] indexing keeps addrspace(3) -> ds_load
                    const float v00 = s_patch[r0 + xl];
                    const float v01 = s_patch[r0 + xh];
                    const float v10 = s_patch[r1 + xl];
                    const float v11 = s_patch[r1 + xh];
                    const float hy = 1.0f - fy, hx = 1.0f - fx;
                    const float val = hy * (hx * v00 + fx * v01)
                                    + fy * (hx * v10 + fx * v11);
                    acc += (vy && vx) ? val : 0.0f;
                }
            }
        } else {
            const float* gch = feat + chan_base + (size_t)c * (size_t)H * (size_t)W;
#pragma unroll
            for (int sy = 0; sy < SRD; ++sy) {
                const float gy = (float)ph + ((float)sy + 0.5f) * 0.5f;
                const float yv = y1s + gy * bin_h;
                const bool  vy = (yv >= -1.0f) && (yv <= (float)H);
                const float yc = fminf(fmaxf(yv, 0.0f), fH1);
                const int   yl = (int)floorf(yc);
                const int   yh = min(yl + 1, H - 1);
                const float fy = yc - (float)yl;
                const size_t r0 = (size_t)yl * (size_t)W;
                const size_t r1 = (size_t)yh * (size_t)W;
#pragma unroll
                for (int sx = 0; sx < SRD; ++sx) {
                    const float gx = (float)pw + ((float)sx + 0.5f) * 0.5f;
                    const float xv = x1s + gx * bin_w;
                    const bool  vx = (xv >= -1.0f) && (xv <= (float)W);
                    const float xc = fminf(fmaxf(xv, 0.0f), fW1);
                    const int   xl = (int)floorf(xc);
                    const int   xh = min(xl + 1, W - 1);
                    const float fx = xc - (float)xl;
                    const float v00 = gch[r0 + xl];
                    const float v01 = gch[r0 + xh];
                    const float v10 = gch[r1 + xl];
                    const float v11 = gch[r1 + xh];
                    const float hy = 1.0f - fy, hx = 1.0f - fx;
                    const float val = hy * (hx * v00 + fx * v01)
                                    + fy * (hx * v10 + fx * v11);
                    acc += (vy && vx) ? val : 0.0f;
                }
            }
        }
        out[((size_t)r * CC + (size_t)(c0 + c)) * (OUTD * OUTD) + bin] = acc * 0.25f;
    }
}

extern "C" void kernel_launch(void* const* d_in, const int* in_sizes, int n_in,
                              void* d_out, int out_size, void* d_ws, size_t ws_size,
                              hipStream_t stream) {
    (void)in_sizes; (void)n_in; (void)out_size; (void)d_ws; (void)ws_size;
    const float* x0    = (const float*)d_in[0];
    const float* x1    = (const float*)d_in[1];
    const float* x2    = (const float*)d_in[2];
    const float* x3    = (const float*)d_in[3];
    const float* boxes = (const float*)d_in[4];
    float* out = (float*)d_out;

    dim3 grid(BN * NN, CC / CPB);   // (512, 32)
    dim3 block(TPB);                // 256 threads = 8 wave32
    roi_align_fpn_kernel<<<grid, block, 0, stream>>>(x0, x1, x2, x3, boxes, out);
}